// KPConvLayer_69320772158013
// MI455X (gfx1250) — compile-verified
//
#include <hip/hip_runtime.h>

#define NPTS 40000
#define MPTS 40000
#define EDG  500000
#define FCH  32
#define CCH  64
#define KPT  15
#define KF   (KPT * FCH)        // 480
#define LDS_STRIDE 482          // words; 482%64=34 -> conflict-free rows, 8B-aligned rows

typedef float v2f __attribute__((ext_vector_type(2)));
typedef float v8f __attribute__((ext_vector_type(8)));

// ---------------------------------------------------------------------------
// Kernel 1: CSR offsets from sorted segment_ids (first index with seg >= m)
// ---------------------------------------------------------------------------
__global__ __launch_bounds__(256) void kp_seg_offsets(const int* __restrict__ seg,
                                                      int* __restrict__ off) {
  int m = blockIdx.x * blockDim.x + threadIdx.x;
  if (m > MPTS) return;
  int lo = 0, hi = EDG;
  while (lo < hi) {
    int mid = (lo + hi) >> 1;
    if (seg[mid] < m) lo = mid + 1; else hi = mid;
  }
  off[m] = lo;
}

// ---------------------------------------------------------------------------
// Kernel 2 (fused): aggregation -> LDS -> WMMA GEMM, one block per 16 outputs.
//
// Phase A: 16 waves; wave w aggregates output point m0+w. Lane = feature
//   channel (F==32 == wave32). Edge scalars are wave-uniform -> SGPR loads.
//   acc[k][lane] written to LDS row w (padded stride).
// Phase B: after barrier, waves 0..3 each compute a 16x16 tile of
//   out[m0:m0+16, 16w:16w+16] = lds_A[16,480] @ kv[480,64] via
//   V_WMMA_F32_16X16X4_F32 (120 WMMAs / wave). A from LDS, B from L2.
//
// f32 WMMA VGPR layouts (ISA 7.12.2):
//   A 16x4 : lanes 0-15 -> K {0,1}, lanes 16-31 -> K {2,3}; M = lane&15
//   B 4x16 : same K split; N = lane&15
//   D 16x16: VGPR i -> M = i (lanes 0-15) / i+8 (lanes 16-31); N = lane&15
// ---------------------------------------------------------------------------
__global__ __launch_bounds__(512) void kp_fused(
    const float* __restrict__ points, const float* __restrict__ features,
    const float* __restrict__ outp,   const int*   __restrict__ nbr,
    const int*   __restrict__ off,    const float* __restrict__ kpts,
    const float* __restrict__ kv,     float* __restrict__ out)
{
  __shared__ float smem[16 * LDS_STRIDE];   // 30848 B

  int lane = threadIdx.x & 31;
  int wv   = threadIdx.x >> 5;              // 0..15
  int m0   = blockIdx.x * 16;

  // ---------------- Phase A: ragged aggregation into LDS -----------------
  {
    int m = m0 + wv;
    float ox = outp[3 * m + 0];
    float oy = outp[3 * m + 1];
    float oz = outp[3 * m + 2];

    float kx[KPT], ky[KPT], kz[KPT];
#pragma unroll
    for (int k = 0; k < KPT; ++k) {
      kx[k] = kpts[3 * k + 0];
      ky[k] = kpts[3 * k + 1];
      kz[k] = kpts[3 * k + 2];
    }

    float acc[KPT];
#pragma unroll
    for (int k = 0; k < KPT; ++k) acc[k] = 0.0f;

    int e0 = off[m];
    int e1 = off[m + 1];
    const float inv_ext = 1.0f / 0.6f;

    for (int e = e0; e < e1; ++e) {
      int nb = __builtin_amdgcn_readfirstlane(nbr[e]);   // wave-uniform
      float rx = points[3 * nb + 0] - ox;
      float ry = points[3 * nb + 1] - oy;
      float rz = points[3 * nb + 2] - oz;
      float fv = features[(size_t)nb * FCH + lane];      // coalesced 128B
#pragma unroll
      for (int k = 0; k < KPT; ++k) {
        float dx = rx - kx[k], dy = ry - ky[k], dz = rz - kz[k];
        float sq = fmaf(dx, dx, fmaf(dy, dy, dz * dz));
        float w  = fmaxf(1.0f - sqrtf(sq) * inv_ext, 0.0f);
        acc[k] = fmaf(w, fv, acc[k]);
      }
    }

    float* row = smem + wv * LDS_STRIDE;
#pragma unroll
    for (int k = 0; k < KPT; ++k) row[k * FCH + lane] = acc[k];
  }

  __syncthreads();

  // ---------------- Phase B: 16x64 GEMM tile via f32 WMMA ----------------
  if (wv < 4) {
    int n0   = wv * 16;
    int half = lane >> 4;        // 0: K+{0,1}, 1: K+{2,3}
    int l15  = lane & 15;

    const float* arow = smem + l15 * LDS_STRIDE + 2 * half;    // LDS, 8B-aligned
    const float* bcol = kv + (size_t)(2 * half) * CCH + n0 + l15;

    v8f acc = {};
#pragma unroll 4
    for (int kk = 0; kk < KF; kk += 4) {
      v2f a, b;
      a.x = arow[kk];                        // K = kk + 2*half
      a.y = arow[kk + 1];                    // K = kk + 2*half + 1
      b.x = bcol[(size_t)kk * CCH];
      b.y = bcol[(size_t)(kk + 1) * CCH];
      acc = __builtin_amdgcn_wmma_f32_16x16x4_f32(
          /*neg_a=*/false, a, /*neg_b=*/false, b,
          /*c_mod=*/(short)0, acc, /*reuse_a=*/false, /*reuse_b=*/false);
    }

    float* orow = out + (size_t)(m0 + half * 8) * CCH + n0 + l15;
#pragma unroll
    for (int i = 0; i < 8; ++i) orow[(size_t)i * CCH] = acc[i];
  }
}

// ---------------------------------------------------------------------------
extern "C" void kernel_launch(void* const* d_in, const int* in_sizes, int n_in,
                              void* d_out, int out_size, void* d_ws, size_t ws_size,
                              hipStream_t stream) {
  const float* points   = (const float*)d_in[0];   // [N,3]
  const float* features = (const float*)d_in[1];   // [N,32]
  const float* outp     = (const float*)d_in[2];   // [M,3]
  const int*   nbr      = (const int*)  d_in[3];   // [E]
  const int*   seg      = (const int*)  d_in[4];   // [E] sorted
  const float* kpts     = (const float*)d_in[5];   // [K,3]
  const float* kv       = (const float*)d_in[6];   // [K,32,64] == W[480,64]
  float* out = (float*)d_out;                      // [M,64]

  int* off = (int*)d_ws;                           // (M+1) ints only

  kp_seg_offsets<<<(MPTS + 1 + 255) / 256, 256, 0, stream>>>(seg, off);

  // 16 output points per block (16 waves of 32)
  kp_fused<<<MPTS / 16, 512, 0, stream>>>(points, features, outp, nbr, off,
                                          kpts, kv, out);
}